// BaseGCN_6725918785568
// MI455X (gfx1250) — compile-verified
//
#include <hip/hip_runtime.h>
#include <hip/hip_bf16.h>

// ---------------------------------------------------------------------------
// 2-layer GCN on bipartite user-item graph (PyG GCNConv semantics).
// N = 150000 nodes, D = 64, E = 2M undirected edges + self loops.
//   deg -> dinv = rsqrt(deg)
//   per layer: h = x@W (WMMA f32 16x16x4), agg[col] += h[row]*dinv[row]*dinv[col]
//              x' = relu(agg + h*dinv^2 (self loop) + b)
// Working set (~150MB) fits MI455X's 192MB L2 -> scatter runs at L2 bandwidth.
// ---------------------------------------------------------------------------

#define NUM_USERS 100000
#define NUM_ITEMS 50000
#define NNODES    (NUM_USERS + NUM_ITEMS)   // 150000
#define DIM       64
#define ND        (NNODES * DIM)            // 9,600,000 floats

typedef float v2f __attribute__((ext_vector_type(2)));
typedef float v8f __attribute__((ext_vector_type(8)));

// ---------------------------------------------------------------------------
// x = concat(users_w, items_w); deg = 1 (self loops)
// ---------------------------------------------------------------------------
__global__ void init_x_deg(const float* __restrict__ uw, const float* __restrict__ iw,
                           float* __restrict__ x, float* __restrict__ deg) {
    int i = blockIdx.x * blockDim.x + threadIdx.x;
    if (i < ND) x[i] = (i < NUM_USERS * DIM) ? uw[i] : iw[i - NUM_USERS * DIM];
    if (i < NNODES) deg[i] = 1.0f;
}

// deg accumulation from undirected edges (each edge adds 1 to both endpoints)
__global__ void accum_deg(const int* __restrict__ eu, const int* __restrict__ ei,
                          float* __restrict__ deg, int E) {
    int e = blockIdx.x * blockDim.x + threadIdx.x;
    if (e >= E) return;
    unsafeAtomicAdd(&deg[eu[e]], 1.0f);
    unsafeAtomicAdd(&deg[ei[e] + NUM_USERS], 1.0f);
}

// dinv = rsqrt(deg), in place (deg >= 1 always due to self loops)
__global__ void finish_dinv(float* __restrict__ deg) {
    int i = blockIdx.x * blockDim.x + threadIdx.x;
    if (i < NNODES) deg[i] = __frsqrt_rn(deg[i]);
}

__global__ void zero_buf(float* __restrict__ p, int n) {
    int i = blockIdx.x * blockDim.x + threadIdx.x;
    if (i < n) p[i] = 0.0f;
}

// ---------------------------------------------------------------------------
// H[N,64] = X[N,64] @ W[64,64] using V_WMMA_F32_16X16X4_F32 (wave32).
// One wave computes a 16-row strip x all 64 cols (4 accumulator tiles).
// A 16x4 f32 frag: lanes 0-15 -> K=k0..k0+1, lanes 16-31 -> K=k0+2..k0+3.
// B 4x16 f32 frag: symmetric (K in VGPR/lane-half, N across lanes 0-15).
// C/D 16x16 f32: VGPR r -> M=r (lanes 0-15) / M=r+8 (lanes 16-31), N=lane&15.
// ---------------------------------------------------------------------------
__global__ void __launch_bounds__(256)
gemm64_wmma(const float* __restrict__ X, const float* __restrict__ W,
            float* __restrict__ H, int nrows) {
    __shared__ float Ws[DIM * DIM];               // 16 KB of 320 KB LDS
    for (int i = threadIdx.x; i < DIM * DIM; i += blockDim.x) Ws[i] = W[i];
    __syncthreads();

    const int lane  = threadIdx.x & 31;
    const int wave  = blockIdx.x * (blockDim.x >> 5) + (threadIdx.x >> 5);
    const int m0    = wave * 16;
    if (m0 >= nrows) return;                      // wave-uniform: EXEC stays all-1s

    const int mrow  = m0 + (lane & 15);           // A-frag row for this lane
    const int khalf = (lane >> 4) << 1;           // 0 for lanes 0-15, 2 for 16-31
    const int bl    = lane & 15;                  // B-frag column within tile

    v8f c0 = {}, c1 = {}, c2 = {}, c3 = {};

    for (int k0 = 0; k0 < DIM; k0 += 4) {
        v2f a;
        a.x = X[mrow * DIM + k0 + khalf + 0];
        a.y = X[mrow * DIM + k0 + khalf + 1];

        const float* w0 = &Ws[(k0 + khalf + 0) * DIM];
        const float* w1 = &Ws[(k0 + khalf + 1) * DIM];
        v2f b0, b1, b2, b3;
        b0.x = w0[bl +  0]; b0.y = w1[bl +  0];
        b1.x = w0[bl + 16]; b1.y = w1[bl + 16];
        b2.x = w0[bl + 32]; b2.y = w1[bl + 32];
        b3.x = w0[bl + 48]; b3.y = w1[bl + 48];

        c0 = __builtin_amdgcn_wmma_f32_16x16x4_f32(false, a, false, b0, (short)0, c0, false, false);
        c1 = __builtin_amdgcn_wmma_f32_16x16x4_f32(false, a, false, b1, (short)0, c1, false, false);
        c2 = __builtin_amdgcn_wmma_f32_16x16x4_f32(false, a, false, b2, (short)0, c2, false, false);
        c3 = __builtin_amdgcn_wmma_f32_16x16x4_f32(false, a, false, b3, (short)0, c3, false, false);
    }

    const int mr = m0 + ((lane >> 4) << 3);       // M = r (+8 for upper lane half)
    const int nc = lane & 15;
    #pragma unroll
    for (int r = 0; r < 8; ++r) {
        H[(mr + r) * DIM + nc +  0] = c0[r];
        H[(mr + r) * DIM + nc + 16] = c1[r];
        H[(mr + r) * DIM + nc + 32] = c2[r];
        H[(mr + r) * DIM + nc + 48] = c3[r];
    }
}

// ---------------------------------------------------------------------------
// Edge aggregation: one wave per undirected edge, both directions at once.
// Lane l handles feature elements l and l+32 (two contiguous 128B half-rows).
// agg[v] += h[u]*nrm ; agg[u] += h[v]*nrm, nrm = dinv[u]*dinv[v].
// Hardware f32 atomics; all traffic should stay resident in 192MB L2.
// ---------------------------------------------------------------------------
__global__ void __launch_bounds__(256)
scatter_edges(const int* __restrict__ eu, const int* __restrict__ ei,
              const float* __restrict__ h, const float* __restrict__ dinv,
              float* __restrict__ agg, int E) {
    int t    = blockIdx.x * blockDim.x + threadIdx.x;
    int e    = t >> 5;
    int lane = t & 31;
    if (e >= E) return;

    int u = eu[e];
    int v = ei[e] + NUM_USERS;
    float nrm = dinv[u] * dinv[v];

    int ub = u * DIM, vb = v * DIM;
    float hu0 = h[ub + lane];
    float hu1 = h[ub + lane + 32];
    float hv0 = h[vb + lane];
    float hv1 = h[vb + lane + 32];

    unsafeAtomicAdd(&agg[vb + lane],      hu0 * nrm);
    unsafeAtomicAdd(&agg[vb + lane + 32], hu1 * nrm);
    unsafeAtomicAdd(&agg[ub + lane],      hv0 * nrm);
    unsafeAtomicAdd(&agg[ub + lane + 32], hv1 * nrm);
}

// x' = relu(agg + h*dinv^2 + b)   (self-loop message folded in here)
__global__ void bias_relu(const float* __restrict__ agg, const float* __restrict__ h,
                          const float* __restrict__ dinv, const float* __restrict__ b,
                          float* __restrict__ x) {
    int i = blockIdx.x * blockDim.x + threadIdx.x;
    if (i >= ND) return;
    int n = i >> 6, d = i & 63;
    float di = dinv[n];
    float v  = agg[i] + h[i] * di * di + b[d];
    x[i] = fmaxf(v, 0.0f);
}

// d_out = [x_users | users_w | x_items | items_w]
__global__ void write_out(const float* __restrict__ x, const float* __restrict__ uw,
                          const float* __restrict__ iw, float* __restrict__ out) {
    int i = blockIdx.x * blockDim.x + threadIdx.x;
    const int U = NUM_USERS * DIM, I = NUM_ITEMS * DIM;
    if (i < U)                 out[i] = x[i];
    else if (i < 2 * U)        out[i] = uw[i - U];
    else if (i < 2 * U + I)    out[i] = x[U + (i - 2 * U)];
    else if (i < 2 * U + 2*I)  out[i] = iw[i - 2 * U - I];
}

// ---------------------------------------------------------------------------
extern "C" void kernel_launch(void* const* d_in, const int* in_sizes, int n_in,
                              void* d_out, int out_size, void* d_ws, size_t ws_size,
                              hipStream_t stream) {
    const int*   edge = (const int*)d_in[0];
    const int    E    = in_sizes[0] / 2;
    const int*   eu   = edge;          // edge_index[0] = user ids
    const int*   ei   = edge + E;      // edge_index[1] = item ids
    const float* uw   = (const float*)d_in[1];
    const float* iw   = (const float*)d_in[2];
    const float* W1   = (const float*)d_in[3];
    const float* b1   = (const float*)d_in[4];
    const float* W2   = (const float*)d_in[5];
    const float* b2   = (const float*)d_in[6];
    float*       out  = (float*)d_out;

    float* x    = (float*)d_ws;        // [ND]
    float* h    = x    + (size_t)ND;   // [ND]
    float* agg  = h    + (size_t)ND;   // [ND]
    float* dinv = agg  + (size_t)ND;   // [NNODES] (deg, then rsqrt in place)

    const int B = 256;
    const int gElem  = (ND + B - 1) / B;
    const int gNode  = (NNODES + B - 1) / B;
    const int gEdge  = (E + B - 1) / B;
    const int gScat  = (E * 32 + B - 1) / B;          // wave per edge
    const int gGemm  = (NNODES / 16 + 7) / 8;         // 8 waves/block, 16 rows/wave
    const int gOut   = (2 * ND + B - 1) / B;

    // normalization
    init_x_deg <<<gElem, B, 0, stream>>>(uw, iw, x, dinv);
    accum_deg  <<<gEdge, B, 0, stream>>>(eu, ei, dinv, E);
    finish_dinv<<<gNode, B, 0, stream>>>(dinv);

    // layer 1
    zero_buf     <<<gElem, B, 0, stream>>>(agg, ND);
    gemm64_wmma  <<<gGemm, B, 0, stream>>>(x, W1, h, NNODES);
    scatter_edges<<<gScat, B, 0, stream>>>(eu, ei, h, dinv, agg, E);
    bias_relu    <<<gElem, B, 0, stream>>>(agg, h, dinv, b1, x);

    // layer 2
    zero_buf     <<<gElem, B, 0, stream>>>(agg, ND);
    gemm64_wmma  <<<gGemm, B, 0, stream>>>(x, W2, h, NNODES);
    scatter_edges<<<gScat, B, 0, stream>>>(eu, ei, h, dinv, agg, E);
    bias_relu    <<<gElem, B, 0, stream>>>(agg, h, dinv, b2, x);

    // pack outputs
    write_out<<<gOut, B, 0, stream>>>(x, uw, iw, out);
}